// FermiLayer_29789893165507
// MI455X (gfx1250) — compile-verified
//
#include <hip/hip_runtime.h>

typedef __attribute__((ext_vector_type(2))) float v2f;
typedef __attribute__((ext_vector_type(8))) float v8f;
typedef __attribute__((__vector_size__(16))) int v4i;   // int4, matches builtin param

#define TANH_GAIN  1.5927812698663606f
#define INV_SQRT2  0.70710678118654752f

#if __has_builtin(__builtin_amdgcn_global_load_async_to_lds_b128) && \
    __has_builtin(__builtin_amdgcn_s_wait_asynccnt)
#define HAVE_ASYNC_LDS 1
#else
#define HAVE_ASYNC_LDS 0
#endif

#define AS1 __attribute__((address_space(1)))
#define AS3 __attribute__((address_space(3)))

// Row stride (in floats) for LDS tiles: 68*4 = 272 bytes keeps b128 stores
// 16B-aligned and makes the b64 fragment loads bank-conflict-free.
#define PAD 68

// V_WMMA_F32_16X16X4_F32 : D(16x16,f32) = A(16x4,f32) x B(4x16,f32) + C
static __device__ __forceinline__ v8f wmma4(v2f a, v2f b, v8f c) {
  return __builtin_amdgcn_wmma_f32_16x16x4_f32(
      false, a, false, b, (short)0, c, false, false);
}

// Stage a 64x64 f32 tile (global row stride = gstride floats) into LDS.
// 128 threads * 8 passes * 16B. Uses CDNA5 async global->LDS DMA when present.
static __device__ __forceinline__ void stage_tile(
    const float* __restrict__ g, int gstride, float (*Xs)[PAD], int tid) {
#if HAVE_ASYNC_LDS
  #pragma unroll
  for (int p = 0; p < 8; ++p) {
    int lin = p * 128 + tid;
    int row = lin >> 4, c4 = (lin & 15) * 4;
    __builtin_amdgcn_global_load_async_to_lds_b128(
        (AS1 v4i*)(AS1 void*)(g + (size_t)row * gstride + c4),
        (AS3 v4i*)(AS3 void*)&Xs[row][c4], 0, 0);
  }
  __builtin_amdgcn_s_wait_asynccnt(0);
#else
  #pragma unroll
  for (int p = 0; p < 8; ++p) {
    int lin = p * 128 + tid;
    int row = lin >> 4, c4 = (lin & 15) * 4;
    float4 v = *(const float4*)(g + (size_t)row * gstride + c4);
    Xs[row][c4 + 0] = v.x; Xs[row][c4 + 1] = v.y;
    Xs[row][c4 + 2] = v.z; Xs[row][c4 + 3] = v.w;
  }
#endif
}

// Load a 64x64 weight tile TRANSPOSED into LDS: Wt[n][k] = W[k][n].
// Makes each lane's B fragment {W[kb][n], W[kb+1][n]} a contiguous b64.
static __device__ __forceinline__ void stage_weights_T(
    const float* __restrict__ W, int gstride, float (*Wt)[PAD], int tid) {
  #pragma unroll
  for (int p = 0; p < 8; ++p) {
    int lin = p * 128 + tid;
    int k = lin >> 4, n4 = (lin & 15) * 4;           // k-row, 4 n-columns
    float4 v = *(const float4*)(W + (size_t)k * gstride + n4);
    Wt[n4 + 0][k] = v.x; Wt[n4 + 1][k] = v.y;
    Wt[n4 + 2][k] = v.z; Wt[n4 + 3][k] = v.w;
  }
}

// ---------------------------------------------------------------------------
// Pair stream: per electron tile t (64 partner rows x 64 feats):
//   mean[t][:] = column mean of X              (feeds one_kernel)
//   Y          = (X + gain*tanh(X@Wp + bp)) / sqrt(2)
// ---------------------------------------------------------------------------
__global__ __launch_bounds__(128) void pair_kernel(
    const float* __restrict__ X, const float* __restrict__ Wp,
    const float* __restrict__ bp, float* __restrict__ Y,
    float* __restrict__ mean) {
  __shared__ __align__(16) float Wt[64][PAD];   // transposed weights
  __shared__ __align__(16) float Xs[64][PAD];
  __shared__ float Bs[64];
  const int tid = threadIdx.x;

  stage_weights_T(Wp, 64, Wt, tid);
  if (tid < 64) Bs[tid] = bp[tid];
  __syncthreads();

  const int lane = tid & 31;
  const int wave = tid >> 5;
  const int half = lane >> 4;   // 0: lanes 0-15, 1: lanes 16-31
  const int mr   = lane & 15;   // M index for A, N index for B/C/D
  const int m0   = wave * 16;

  for (int t = blockIdx.x; t < 4096; t += gridDim.x) {
    stage_tile(X + (size_t)t * 4096, 64, Xs, tid);
    __syncthreads();

    // Column means over 64 partner rows.
    if (tid < 64) {
      float s = 0.f;
      #pragma unroll
      for (int r = 0; r < 64; ++r) s += Xs[r][tid];
      mean[(size_t)t * 64 + tid] = s * (1.0f / 64.0f);
    }

    v8f acc0 = {}, acc1 = {}, acc2 = {}, acc3 = {};
    #pragma unroll
    for (int k4 = 0; k4 < 16; ++k4) {
      const int kb = k4 * 4 + half * 2;                  // ISA 16x4 A layout
      v2f a  = *(const v2f*)&Xs[m0 + mr][kb];            // one ds_load_b64
      v2f b0 = *(const v2f*)&Wt[ 0 + mr][kb];
      v2f b1 = *(const v2f*)&Wt[16 + mr][kb];
      v2f b2 = *(const v2f*)&Wt[32 + mr][kb];
      v2f b3 = *(const v2f*)&Wt[48 + mr][kb];
      acc0 = wmma4(a, b0, acc0);
      acc1 = wmma4(a, b1, acc1);
      acc2 = wmma4(a, b2, acc2);
      acc3 = wmma4(a, b3, acc3);
    }

    float* Yg = Y + (size_t)t * 4096;
    auto epi = [&](v8f acc, int tn) {
      const int n = tn * 16 + mr;
      #pragma unroll
      for (int r = 0; r < 8; ++r) {
        const int m = m0 + r + half * 8;                 // C/D VGPR layout
        float y = TANH_GAIN * tanhf(acc[r] + Bs[n]);
        Yg[m * 64 + n] = (Xs[m][n] + y) * INV_SQRT2;
      }
    };
    epi(acc0, 0); epi(acc1, 1); epi(acc2, 2); epi(acc3, 3);
    __syncthreads();
  }
}

// ---------------------------------------------------------------------------
__global__ __launch_bounds__(256) void gmean_kernel(
    const float* __restrict__ h_one, float* __restrict__ gmean) {
  const int j = blockIdx.x;       // 0..63 (group = 2*g + spin)
  const int d = threadIdx.x;      // 0..255
  const float* p = h_one + (size_t)j * 64 * 256 + d;
  float s = 0.f;
  #pragma unroll 8
  for (int r = 0; r < 64; ++r) s += p[r * 256];
  gmean[j * 256 + d] = s * (1.0f / 64.0f);
}

__global__ __launch_bounds__(256) void gnew_kernel(
    const float* __restrict__ gmean, const float* __restrict__ Wg,
    float* __restrict__ gnew) {
  const int j = blockIdx.x;
  const int n = threadIdx.x;
  const float* own = gmean + j * 256;
  const float* oth = gmean + (j ^ 1) * 256;
  float s = 0.f;
  for (int k = 0; k < 256; ++k) s += own[k] * Wg[k * 256 + n];
  for (int k = 0; k < 256; ++k) s += oth[k] * Wg[(256 + k) * 256 + n];
  gnew[j * 256 + n] = s;
}

// ---------------------------------------------------------------------------
__global__ __launch_bounds__(128) void one_kernel(
    const float* __restrict__ h_one, const float* __restrict__ mean_s,
    const float* __restrict__ mean_d, const float* __restrict__ W1,
    const float* __restrict__ b1, const float* __restrict__ gnew,
    float* __restrict__ out) {
  __shared__ __align__(16) float Xs[64][PAD];
  __shared__ __align__(16) float Wt[64][PAD];
  __shared__ float Bs[64];
  __shared__ float Gs[64];
  const int tid = threadIdx.x;
  const int bm = blockIdx.x;      // 0..63 : row tile AND spin group
  const int bn = blockIdx.y;      // 0..3  : column tile of 64

  if (tid < 64) {
    Bs[tid] = b1[bn * 64 + tid];
    Gs[tid] = gnew[bm * 256 + bn * 64 + tid];
  }

  const int lane = tid & 31;
  const int wave = tid >> 5;
  const int half = lane >> 4;
  const int mr   = lane & 15;
  const int m0   = wave * 16;

  v8f acc0 = {}, acc1 = {}, acc2 = {}, acc3 = {};

  for (int kc = 0; kc < 6; ++kc) {
    const float* xsrc;
    int xstride;
    if (kc < 4)       { xsrc = h_one  + (size_t)(bm * 64) * 256 + kc * 64; xstride = 256; }
    else if (kc == 4) { xsrc = mean_s + (size_t)(bm * 64) * 64;            xstride = 64;  }
    else              { xsrc = mean_d + (size_t)(bm * 64) * 64;            xstride = 64;  }
    stage_tile(xsrc, xstride, Xs, tid);
    stage_weights_T(W1 + (size_t)(kc * 64) * 256 + bn * 64, 256, Wt, tid);
    __syncthreads();

    #pragma unroll
    for (int k4 = 0; k4 < 16; ++k4) {
      const int kb = k4 * 4 + half * 2;
      v2f a  = *(const v2f*)&Xs[m0 + mr][kb];
      v2f b0 = *(const v2f*)&Wt[ 0 + mr][kb];
      v2f b1 = *(const v2f*)&Wt[16 + mr][kb];
      v2f b2 = *(const v2f*)&Wt[32 + mr][kb];
      v2f b3 = *(const v2f*)&Wt[48 + mr][kb];
      acc0 = wmma4(a, b0, acc0);
      acc1 = wmma4(a, b1, acc1);
      acc2 = wmma4(a, b2, acc2);
      acc3 = wmma4(a, b3, acc3);
    }
    __syncthreads();
  }

  auto epi = [&](v8f acc, int tn) {
    const int n    = tn * 16 + mr;
    const int gcol = bn * 64 + n;
    #pragma unroll
    for (int r = 0; r < 8; ++r) {
      const int gm = bm * 64 + m0 + r + half * 8;
      float v = (acc[r] + Bs[n] + Gs[n]) * INV_SQRT2;
      float y = TANH_GAIN * tanhf(v);
      out[(size_t)gm * 256 + gcol] =
          (h_one[(size_t)gm * 256 + gcol] + y) * INV_SQRT2;
    }
  };
  epi(acc0, 0); epi(acc1, 1); epi(acc2, 2); epi(acc3, 3);
}

// ---------------------------------------------------------------------------
extern "C" void kernel_launch(void* const* d_in, const int* in_sizes, int n_in,
                              void* d_out, int out_size, void* d_ws, size_t ws_size,
                              hipStream_t stream) {
  (void)in_sizes; (void)n_in; (void)out_size; (void)ws_size;
  const float* h_one      = (const float*)d_in[0];
  const float* h_two_same = (const float*)d_in[1];
  const float* h_two_diff = (const float*)d_in[2];
  const float* W1         = (const float*)d_in[3];
  const float* b1         = (const float*)d_in[4];
  const float* Wg         = (const float*)d_in[5];
  const float* Wp_same    = (const float*)d_in[6];
  const float* bp_same    = (const float*)d_in[7];
  const float* Wp_diff    = (const float*)d_in[8];
  const float* bp_diff    = (const float*)d_in[9];

  float* out_one  = (float*)d_out;                       // 4096 x 256
  float* out_same = out_one  + (size_t)4096 * 256;       // 262144 x 64
  float* out_diff = out_same + (size_t)262144 * 64;      // 262144 x 64

  float* ws     = (float*)d_ws;
  float* mean_s = ws;                                    // 4096 x 64
  float* mean_d = mean_s + (size_t)4096 * 64;            // 4096 x 64
  float* gmean  = mean_d + (size_t)4096 * 64;            // 64 x 256
  float* gnew   = gmean  + 64 * 256;                     // 64 x 256

  pair_kernel<<<dim3(1024), 128, 0, stream>>>(h_two_same, Wp_same, bp_same,
                                              out_same, mean_s);
  pair_kernel<<<dim3(1024), 128, 0, stream>>>(h_two_diff, Wp_diff, bp_diff,
                                              out_diff, mean_d);
  gmean_kernel<<<64, 256, 0, stream>>>(h_one, gmean);
  gnew_kernel<<<64, 256, 0, stream>>>(gmean, Wg, gnew);
  one_kernel<<<dim3(64, 4), 128, 0, stream>>>(h_one, mean_s, mean_d,
                                              W1, b1, gnew, out_one);
}